// ROIBoxHead_4423816315545
// MI455X (gfx1250) — compile-verified
//
#include <hip/hip_runtime.h>
#include <hip/hip_bf16.h>

// ---------------------------------------------------------------------------
// ROI box head post-processing for MI455X (gfx1250, wave32).
//   B=2 images, N=1000 proposals, C=81 classes, K=100 detections/img.
// Pipeline: softmax -> bbox decode+clip -> per-class stable sort -> greedy NMS
//           -> per-class top-100 candidates -> global top-100 + labels.
// CDNA5 features: wave32 shuffle reductions; LDS-resident NMS; Tensor Data
// Mover DMAs (1-D score row, 1-D proposal row, 2-D strided regression tile)
// with s_wait_tensorcnt completion.
// ---------------------------------------------------------------------------

#define BN      1000
#define NCLS    81
#define FG      80          // C-1
#define KDET    100
#define NMS_TH  0.5f
#define SCORE_TH 0.05f
#define XFORM_CLIP 4.135166556742356f   // log(1000/16)
#define WCLIP 1332.0f       // W_IMG - 1
#define HCLIP 799.0f        // H_IMG - 1
#define CAND_STRIDE 8       // words per candidate record

typedef unsigned int u32;
typedef u32 u32x4 __attribute__((ext_vector_type(4)));
typedef u32 u32x8 __attribute__((ext_vector_type(8)));

// ---------------------------------------------------------------------------
// TDM issue helper: D# group0 (count=1, lds_addr, global_addr, type=2) +
// caller-supplied group1 (dims/strides). 2-SGPR-group form (<=2-D tensors).
// ---------------------------------------------------------------------------
static __device__ __forceinline__ void tdm_load_to_lds(
    const void* gptr, u32 ldsOff,
    u32 w0, u32 w1, u32 w2, u32 w3, u32 w4, u32 w5, u32 w6, u32 w7)
{
    const unsigned long long ga = (unsigned long long)(uintptr_t)gptr;
    u32x4 g0;
    g0[0] = 1u;                                            // count=1, user mode
    g0[1] = ldsOff;                                        // lds_addr (bytes)
    g0[2] = (u32)ga;                                       // global_addr[31:0]
    g0[3] = (u32)((ga >> 32) & 0x1FFFFFFull) | (2u << 30); // [56:32] | type=2
    u32x8 g1;
    g1[0] = w0; g1[1] = w1; g1[2] = w2; g1[3] = w3;
    g1[4] = w4; g1[5] = w5; g1[6] = w6; g1[7] = w7;
    asm volatile("tensor_load_to_lds %0, %1" :: "s"(g0), "s"(g1) : "memory");
}

// ---------------------------------------------------------------------------
// Kernel 1: softmax over 81 logits, one wave32 per (b, n) row.
// Writes fg probs class-major: scores[((b*FG + (c-1)) * BN) + n].
// ---------------------------------------------------------------------------
__global__ __launch_bounds__(256)
void k1_softmax(const float* __restrict__ logits, float* __restrict__ scores)
{
    const int w    = blockIdx.x * 8 + (threadIdx.x >> 5);   // row id in [0, 2*BN)
    const int lane = threadIdx.x & 31;
    if (w >= 2 * BN) return;
    const int b = w / BN;
    const int n = w % BN;

    const float* row = logits + (size_t)w * NCLS;
    const float l0 = row[lane];            // classes 0..31
    const float l1 = row[lane + 32];       // classes 32..63
    const bool  h2 = (lane + 64) < NCLS;   // classes 64..80 (lane<=16)
    const float l2 = h2 ? row[lane + 64] : -1e30f;

    float m = fmaxf(l0, fmaxf(l1, l2));
    #pragma unroll
    for (int o = 16; o > 0; o >>= 1) m = fmaxf(m, __shfl_xor(m, o, 32));

    const float e0 = expf(l0 - m);
    const float e1 = expf(l1 - m);
    const float e2 = h2 ? expf(l2 - m) : 0.0f;
    float s = e0 + e1 + e2;
    #pragma unroll
    for (int o = 16; o > 0; o >>= 1) s += __shfl_xor(s, o, 32);

    if (lane >= 1)
        scores[((size_t)(b * FG + (lane - 1))) * BN + n] = e0 / s;
    scores[((size_t)(b * FG + (lane + 31))) * BN + n] = e1 / s;
    if (h2)
        scores[((size_t)(b * FG + (lane + 63))) * BN + n] = e2 / s;
}

// ---------------------------------------------------------------------------
// Kernel 2: per (b, class). TDM stages scores / proposals / class-deltas into
// LDS; then stable desc sort (rank), decode+clip in sorted order, greedy NMS,
// per-class top-100 candidate emission. 256 threads (8 waves) per block.
//
// LDS arena (60 KB, float offsets), with phase-based overlays:
//   SC  [0,1000)      raw scores (TDM dst)    -> later post-NMS scores
//   SS  [1000,2000)   sorted scores
//   ORD [2000,3000)   sorted -> original index (int view)
//   BOX [3000,7000)   decoded float4 boxes (16B aligned: 3000*4 % 16 == 0)
//   PR  [7000,11000)  proposal tile (TDM dst) -> later KEPT/SUPP flags
//   RG  [11000,15000) regression tile (TDM dst, 2-D strided)
// ---------------------------------------------------------------------------
#define OFF_SC   0
#define OFF_SS   1000
#define OFF_ORD  2000
#define OFF_BOX  3000
#define OFF_PR   7000
#define OFF_RG   11000
#define OFF_KEPT 7000      // overlays PR (dead after decode)
#define OFF_SUPP 8000      // overlays PR+1000

__global__ __launch_bounds__(256)
void k2_sort_nms(const float*  __restrict__ reg,      // [B, N, C*4]
                 const float*  __restrict__ props,    // [B, N, 4]
                 const float*  __restrict__ scores,   // [B*FG, N] class-major
                 float*        __restrict__ cand)     // [B*FG, KDET, 8]
{
    const int bc  = blockIdx.x;           // b*FG + c
    const int b   = bc / FG;
    const int c   = bc % FG;
    const int tid = threadIdx.x;

    __shared__ __align__(16) float smem[15000];
    int* smemi = (int*)smem;

    // ---- TDM: three DMAs, wave 0 issues, single tensorcnt wait.
    if (tid < 32) {
        // (a) score row: 1-D tile of 1000 f32 (contiguous)
        tdm_load_to_lds(scores + (size_t)bc * BN,
                        (u32)(uintptr_t)(void*)&smem[OFF_SC],
                        (2u << 16),                    // data_size = 4B
                        ((u32)BN) << 16,               // tensor_dim0 = 1000
                        (1u << 16),                    // tensor_dim1 = 1
                        ((u32)BN) << 16,               // tile_dim0 = 1000
                        0u,                            // tile_dim1/2 = 0 (1-D)
                        (u32)BN, 0u, 0u);              // dim0_stride (unused)
        // (b) proposal row for image b: 1-D tile of 4000 f32 (contiguous)
        tdm_load_to_lds(props + (size_t)b * BN * 4,
                        (u32)(uintptr_t)(void*)&smem[OFF_PR],
                        (2u << 16),
                        ((u32)(BN * 4)) << 16,         // tensor_dim0 = 4000
                        (1u << 16),                    // tensor_dim1 = 1
                        ((u32)(BN * 4)) << 16,         // tile_dim0 = 4000
                        0u,
                        (u32)(BN * 4), 0u, 0u);
        // (c) regression deltas for class c+1: 2-D tile 4 x 1000,
        //     row stride 324 f32 -> packed 16 KB in LDS.
        tdm_load_to_lds(reg + (size_t)b * BN * (NCLS * 4) + (size_t)(c + 1) * 4,
                        (u32)(uintptr_t)(void*)&smem[OFF_RG],
                        (2u << 16),
                        (4u << 16),                    // tensor_dim0 = 4
                        ((u32)BN) << 16,               // tensor_dim1 = 1000
                        (4u << 16),                    // tile_dim0 = 4
                        (u32)BN,                       // tile_dim1 = 1000
                        (u32)(NCLS * 4), 0u, 0u);      // dim0_stride = 324
        __builtin_amdgcn_s_wait_tensorcnt(0);
    }
    __syncthreads();

    // ---- stable descending argsort via rank counting (ties -> lower index)
    for (int i = tid; i < BN; i += 256) {
        const float si = smem[OFF_SC + i];
        int r = 0;
        for (int j = 0; j < BN; ++j) {
            const float sj = smem[OFF_SC + j];
            r += (sj > si) || ((sj == si) && (j < i));
        }
        smemi[OFF_ORD + r] = i;
        smem[OFF_SS + r]   = si;
    }
    __syncthreads();

    // ---- decode + clip in sorted order (all inputs now LDS-resident)
    for (int p = tid; p < BN; p += 256) {
        const int n = smemi[OFF_ORD + p];
        const float x1 = smem[OFF_PR + n * 4 + 0];
        const float y1 = smem[OFF_PR + n * 4 + 1];
        const float x2 = smem[OFF_PR + n * 4 + 2];
        const float y2 = smem[OFF_PR + n * 4 + 3];
        const float d0 = smem[OFF_RG + n * 4 + 0];
        const float d1 = smem[OFF_RG + n * 4 + 1];
        const float d2 = smem[OFF_RG + n * 4 + 2];
        const float d3 = smem[OFF_RG + n * 4 + 3];
        const float w  = x2 - x1 + 1.0f;
        const float h  = y2 - y1 + 1.0f;
        const float cx = x1 + 0.5f * w;
        const float cy = y1 + 0.5f * h;
        const float dx = d0 / 10.0f;
        const float dy = d1 / 10.0f;
        const float dw = fminf(d2 / 5.0f, XFORM_CLIP);
        const float dh = fminf(d3 / 5.0f, XFORM_CLIP);
        const float pcx = dx * w + cx;
        const float pcy = dy * h + cy;
        const float pw  = expf(dw) * w;
        const float ph  = expf(dh) * h;
        float bx1 = pcx - 0.5f * pw;
        float by1 = pcy - 0.5f * ph;
        float bx2 = pcx + 0.5f * pw - 1.0f;
        float by2 = pcy + 0.5f * ph - 1.0f;
        bx1 = fminf(fmaxf(bx1, 0.0f), WCLIP);
        by1 = fminf(fmaxf(by1, 0.0f), HCLIP);
        bx2 = fminf(fmaxf(bx2, 0.0f), WCLIP);
        by2 = fminf(fmaxf(by2, 0.0f), HCLIP);
        ((float4*)&smem[OFF_BOX])[p] = make_float4(bx1, by1, bx2, by2);
    }
    __syncthreads();                      // PR/RG tiles dead beyond this point

    for (int i = tid; i < BN; i += 256) smemi[OFF_SUPP + i] = 0;
    __syncthreads();

    // ---- greedy NMS (reference semantics: only kept & valid boxes suppress)
    const float4* boxv = (const float4*)&smem[OFF_BOX];
    for (int i = 0; i < BN; ++i) {
        if (tid == 0)
            smemi[OFF_KEPT + i] = (smem[OFF_SS + i] > SCORE_TH) &&
                                  (smemi[OFF_SUPP + i] == 0);
        __syncthreads();
        if (smemi[OFF_KEPT + i]) {        // uniform branch (LDS read post-barrier)
            const float4 bi = boxv[i];
            const float  ai = fmaxf(bi.z - bi.x + 1.0f, 0.0f) *
                              fmaxf(bi.w - bi.y + 1.0f, 0.0f);
            for (int j = i + 1 + tid; j < BN; j += 256) {
                if (smemi[OFF_SUPP + j]) continue;
                const float4 bj = boxv[j];
                const float  aj = fmaxf(bj.z - bj.x + 1.0f, 0.0f) *
                                  fmaxf(bj.w - bj.y + 1.0f, 0.0f);
                const float ltx = fmaxf(bi.x, bj.x);
                const float lty = fmaxf(bi.y, bj.y);
                const float rbx = fminf(bi.z, bj.z);
                const float rby = fminf(bi.w, bj.w);
                const float iw  = fmaxf(rbx - ltx + 1.0f, 0.0f);
                const float ih  = fmaxf(rby - lty + 1.0f, 0.0f);
                const float inter = iw * ih;
                const float iou = inter / (ai + aj - inter + 1e-12f);
                if (iou > NMS_TH) smemi[OFF_SUPP + j] = 1;
            }
        }
        __syncthreads();
    }

    // post-NMS scores overwrite the raw-score region (no longer needed)
    for (int p = tid; p < BN; p += 256)
        smem[OFF_SC + p] = smemi[OFF_KEPT + p] ? smem[OFF_SS + p] : -1.0f;
    __syncthreads();

    // ---- per-class top-100 candidates ordered by (score desc, position asc).
    // Superset of the global top-100 under identical tie ordering.
    float* candC = cand + (size_t)bc * KDET * CAND_STRIDE;
    for (int p = tid; p < BN; p += 256) {
        const float sp = smem[OFF_SC + p];
        int r = 0;
        for (int j = 0; j < BN; ++j) {
            const float sj = smem[OFF_SC + j];
            r += (sj > sp) || ((sj == sp) && (j < p));
        }
        if (r < KDET) {
            float* rec = candC + (size_t)r * CAND_STRIDE;
            rec[0] = sp;
            ((int*)rec)[1] = c * BN + p;      // flat index in [FG*N] space
            const float4 bp = boxv[p];
            rec[2] = bp.x; rec[3] = bp.y; rec[4] = bp.z; rec[5] = bp.w;
        }
    }
}

// ---------------------------------------------------------------------------
// Kernel 3: global top-100 per image over the 8000 candidates.
// Rank by (score desc, flat index asc) == jax.lax.top_k tie behavior.
// Ranks are unique, so exactly KDET winners fill the output rows.
// ---------------------------------------------------------------------------
__global__ __launch_bounds__(256)
void k3_topk(const float* __restrict__ cand, float* __restrict__ out)
{
    const int M = FG * KDET;                 // 8000 candidates per image
    const int blocksPerImg = (M + 255) / 256;
    const int b = blockIdx.x / blocksPerImg;
    const int i = (blockIdx.x % blocksPerImg) * 256 + threadIdx.x;
    if (i >= M) return;

    const float* cb = cand + (size_t)b * M * CAND_STRIDE;
    const float2 me = *(const float2*)(cb + (size_t)i * CAND_STRIDE);
    const float si = me.x;
    const int   fi = __float_as_int(me.y);

    int r = 0;
    for (int j = 0; j < M; ++j) {
        const float2 cj = *(const float2*)(cb + (size_t)j * CAND_STRIDE);
        const float sj = cj.x;
        const int   fj = __float_as_int(cj.y);
        r += (sj > si) || ((sj == si) && (fj < fi));
    }
    if (r < KDET) {
        float* ob = out;                         // topk_b  [B, K, 4]
        float* os = out + 2 * KDET * 4;          // topk_s  [B, K]
        float* ol = os + 2 * KDET;               // labels  [B, K]
        const float* rec = cb + (size_t)i * CAND_STRIDE;
        const int o = b * KDET + r;
        ob[o * 4 + 0] = rec[2];
        ob[o * 4 + 1] = rec[3];
        ob[o * 4 + 2] = rec[4];
        ob[o * 4 + 3] = rec[5];
        os[o] = si;
        ol[o] = (si > 0.0f) ? (float)(fi / BN + 1) : 0.0f;
    }
}

// ---------------------------------------------------------------------------
extern "C" void kernel_launch(void* const* d_in, const int* in_sizes, int n_in,
                              void* d_out, int out_size, void* d_ws, size_t ws_size,
                              hipStream_t stream)
{
    (void)in_sizes; (void)n_in; (void)out_size; (void)ws_size;
    const float* logits = (const float*)d_in[0];   // [B, N, C]
    const float* reg    = (const float*)d_in[1];   // [B, N, C*4]
    const float* props  = (const float*)d_in[2];   // [B, N, 4]
    float* out = (float*)d_out;

    float* wsScores = (float*)d_ws;                         // B*FG*BN floats
    float* wsCand   = wsScores + (size_t)2 * FG * BN;       // B*FG*KDET*8 floats

    // K1: 2*BN rows, 8 wave32 rows per 256-thread block
    k1_softmax<<<(2 * BN) / 8, 256, 0, stream>>>(logits, wsScores);
    // K2: one block per (b, class); 3 TDM DMAs each
    k2_sort_nms<<<2 * FG, 256, 0, stream>>>(reg, props, wsScores, wsCand);
    // K3: global top-K per image
    const int M = FG * KDET;
    const int blocksPerImg = (M + 255) / 256;
    k3_topk<<<2 * blocksPerImg, 256, 0, stream>>>(wsCand, out);
}